// Char_Segmention_Loss_38680475468305
// MI455X (gfx1250) — compile-verified
//
#include <hip/hip_runtime.h>
#include <stdint.h>

namespace {
constexpr int kB = 512;
constexpr int kC = 27;
constexpr int kNCh = 26;
constexpr int kAttnW = 32;
constexpr int kHMid = 32, kWMid = 128, kHWMid = kHMid * kWMid;   // 4096
constexpr int kHLow = 16, kWLow = 64,  kHWLow = kHLow * kWLow;   // 1024
}

typedef __attribute__((ext_vector_type(4))) unsigned int u32x4_t;
typedef __attribute__((ext_vector_type(8))) int i32x8_t;
typedef __attribute__((ext_vector_type(4))) int i32x4_t;

// Issue a TDM 2D tile load: tile0 contiguous elements (f32) x tile1 rows,
// row stride = stride0_elems elements, into LDS at lds_addr.
__device__ __forceinline__ void tdm_load_2d(uint32_t lds_addr, const void* gptr,
                                            uint32_t tdim0, uint32_t tdim1,
                                            uint32_t tile0, uint32_t tile1,
                                            uint64_t stride0_elems) {
  uint64_t ga = (uint64_t)(uintptr_t)gptr;
  u32x4_t g0;
  g0.x = 1u;                                                    // count=1, user mode
  g0.y = lds_addr;                                              // LDS byte address
  g0.z = (uint32_t)ga;                                          // global_addr[31:0]
  g0.w = (uint32_t)((ga >> 32) & 0x01FFFFFFull) | (2u << 30);   // addr[56:32] | type=2
  i32x8_t g1;
  g1[0] = (int)(2u << 16);                                      // data_size=4B
  g1[1] = (int)((tdim0 & 0xFFFFu) << 16);                       // tensor_dim0 lo
  g1[2] = (int)((tdim0 >> 16) | ((tdim1 & 0xFFFFu) << 16));     // dim0 hi | dim1 lo
  g1[3] = (int)((tdim1 >> 16) | (tile0 << 16));                 // dim1 hi | tile_dim0
  g1[4] = (int)(tile1 & 0xFFFFu);                               // tile_dim1, tile_dim2=0
  g1[5] = (int)(uint32_t)(stride0_elems & 0xFFFFFFFFull);       // dim0_stride lo
  g1[6] = (int)(uint32_t)((stride0_elems >> 32) & 0xFFFFull);   // dim0_stride hi, dim1_stride=0
  g1[7] = 0;
  i32x4_t z4 = {0, 0, 0, 0};
  i32x8_t z8 = {0, 0, 0, 0, 0, 0, 0, 0};
  __builtin_amdgcn_tensor_load_to_lds(g0, g1, z4, z4, z8, 0);
}

// ---------------------------------------------------------------------------
// Kernel A: resized+thresholded attention bitmasks per (b, w_out).
// half-pixel bilinear along W (32 -> 64 / 128), rows >= length-1 cleared.
// ---------------------------------------------------------------------------
__global__ void attn_mask_kernel(const float* __restrict__ attn,
                                 const int* __restrict__ length,
                                 const float* __restrict__ alpha_p,
                                 uint32_t* __restrict__ wmask_low,
                                 uint32_t* __restrict__ wmask_mid) {
  const int tid = blockIdx.x * blockDim.x + threadIdx.x;
  const int total = kB * (kWLow + kWMid);
  if (tid >= total) return;
  int b, w, W;
  uint32_t* out;
  if (tid < kB * kWLow) {
    b = tid / kWLow; w = tid % kWLow; W = kWLow;
    out = wmask_low + b * kWLow + w;
  } else {
    const int t = tid - kB * kWLow;
    b = t / kWMid; w = t % kWMid; W = kWMid;
    out = wmask_mid + b * kWMid + w;
  }
  const float alpha = *alpha_p;
  const int len1 = length[b] - 1;
  const float src = (w + 0.5f) * ((float)kAttnW / (float)W) - 0.5f;
  const float f0 = floorf(src);
  const int i0 = (int)f0;
  const float fr = src - f0;
  const int i0c = i0 < 0 ? 0 : (i0 > kAttnW - 1 ? kAttnW - 1 : i0);
  const int i1 = i0 + 1;
  const int i1c = i1 < 0 ? 0 : (i1 > kAttnW - 1 ? kAttnW - 1 : i1);
  const float* a = attn + (size_t)b * kNCh * kAttnW;
  uint32_t bits = 0;
  for (int c = 0; c < kNCh; ++c) {
    float v = 0.f;
    if (c < len1) v = (1.f - fr) * a[c * kAttnW + i0c] + fr * a[c * kAttnW + i1c];
    bits |= (v > alpha ? 1u : 0u) << c;
  }
  *out = bits;
}

// ---------------------------------------------------------------------------
// Kernel B: fore-mask column bitmasks (mid: direct threshold; low: exact 2x2
// average threshold) + analytic s2[b][c] mask sums.
// ---------------------------------------------------------------------------
__global__ __launch_bounds__(128) void fm_s2_kernel(
    const float* __restrict__ fore, const uint32_t* __restrict__ wmask_low,
    const uint32_t* __restrict__ wmask_mid, uint32_t* __restrict__ fmcol_low,
    uint32_t* __restrict__ fmcol_mid, float* __restrict__ s2_low,
    float* __restrict__ s2_mid) {
  const int b = blockIdx.x;
  const int tid = threadIdx.x;
  __shared__ int ccm[kWMid];
  __shared__ int ccl[kWLow];
  const float* F = fore + (size_t)b * kHMid * kWMid;
  {
    const int w = tid;
    uint32_t bits = 0; int cnt = 0;
    for (int h = 0; h < kHMid; ++h) {
      const int f = F[h * kWMid + w] >= 0.4f;
      bits |= (uint32_t)f << h;
      cnt += f;
    }
    fmcol_mid[b * kWMid + w] = bits;
    ccm[w] = cnt;
  }
  if (tid < kWLow) {
    const int w = tid;
    uint32_t bits = 0; int cnt = 0;
    for (int h = 0; h < kHLow; ++h) {
      const float v = 0.25f * (F[(2 * h) * kWMid + 2 * w] + F[(2 * h) * kWMid + 2 * w + 1] +
                               F[(2 * h + 1) * kWMid + 2 * w] + F[(2 * h + 1) * kWMid + 2 * w + 1]);
      const int f = v >= 0.4f;
      bits |= (uint32_t)f << h;
      cnt += f;
    }
    fmcol_low[b * kWLow + w] = bits;
    ccl[w] = cnt;
  }
  __syncthreads();
  if (tid < kC) {
    const int c = tid;
    float s2m, s2l;
    if (c == 0) {
      int sm = 0, sl = 0;
      for (int w = 0; w < kWMid; ++w) sm += ccm[w];
      for (int w = 0; w < kWLow; ++w) sl += ccl[w];
      s2m = (float)(kHWMid - sm);
      s2l = (float)(kHWLow - sl);
    } else {
      int sm = 0, sl = 0;
      for (int w = 0; w < kWMid; ++w) sm += (int)((wmask_mid[b * kWMid + w] >> (c - 1)) & 1u) * ccm[w];
      for (int w = 0; w < kWLow; ++w) sl += (int)((wmask_low[b * kWLow + w] >> (c - 1)) & 1u) * ccl[w];
      s2m = (float)sm;
      s2l = (float)sl;
    }
    s2_mid[b * kC + c] = s2m;
    s2_low[b * kC + c] = s2l;
  }
}

// ---------------------------------------------------------------------------
// Kernel C: reconstruct mask_low [512][27][16][64] as floats into d_out+2.
// ---------------------------------------------------------------------------
__global__ void mask_low_out_kernel(const uint32_t* __restrict__ wmask_low,
                                    const uint32_t* __restrict__ fmcol_low,
                                    float* __restrict__ out) {
  const int idx = blockIdx.x * blockDim.x + threadIdx.x;
  const int total = kB * kC * kHLow * kWLow;
  if (idx >= total) return;
  const int w = idx & (kWLow - 1);
  const int h = (idx >> 6) & (kHLow - 1);
  const int bc = idx >> 10;
  const int c = bc % kC;
  const int b = bc / kC;
  const uint32_t fm = (fmcol_low[b * kWLow + w] >> h) & 1u;
  float v;
  if (c == 0) v = (float)(1u - fm);
  else v = (float)(fm & ((wmask_low[b * kWLow + w] >> (c - 1)) & 1u));
  out[idx] = v;
}

// ---------------------------------------------------------------------------
// Main kernel: one block per sample. TDM double-buffers 27x256 f32 tiles of
// char_feature into LDS; each thread handles one pixel per chunk: 27-way
// softmax stats -> CE term + per-channel s1/inter accumulation.
// ---------------------------------------------------------------------------
template <int H, int W, int NCHUNK>
__global__ __launch_bounds__(256) void char_loss_kernel(
    const float* __restrict__ cf, const uint32_t* __restrict__ wmask,
    const uint32_t* __restrict__ fmcol, float* __restrict__ s1_out,
    float* __restrict__ inter_out, float* __restrict__ ce_part) {
  constexpr int CHUNK = 256;
  constexpr int HW = H * W;
  static_assert(NCHUNK * CHUNK == HW, "chunking must tile HW exactly");
  const int b = blockIdx.x;
  const int tid = threadIdx.x;

  __shared__ float tile[2][kC * CHUNK];   // 2 x 27KB
  __shared__ float red[2 * kC + 1];       // s1[27], inter[27], ce
  if (tid < 2 * kC + 1) red[tid] = 0.f;

  const uint32_t lds0 = (uint32_t)(uintptr_t)(&tile[0][0]);
  const uint32_t lds1 = (uint32_t)(uintptr_t)(&tile[1][0]);
  const float* gbase = cf + (size_t)b * kC * HW;

  float s1a[kC], ina[kC];
#pragma unroll
  for (int c = 0; c < kC; ++c) { s1a[c] = 0.f; ina[c] = 0.f; }
  float cea = 0.f;

  if (tid < 32) tdm_load_2d(lds0, gbase, HW, kC, CHUNK, kC, (uint64_t)HW);

  for (int ch = 0; ch < NCHUNK; ++ch) {
    if (tid < 32) {
      if (ch + 1 < NCHUNK) {
        tdm_load_2d(((ch + 1) & 1) ? lds1 : lds0,
                    gbase + (size_t)(ch + 1) * CHUNK, HW, kC, CHUNK, kC,
                    (uint64_t)HW);
        __builtin_amdgcn_s_wait_tensorcnt(1);  // chunk `ch` complete (in-order)
      } else {
        __builtin_amdgcn_s_wait_tensorcnt(0);
      }
    }
    __syncthreads();

    const float* L = &tile[ch & 1][tid];
    const int p = ch * CHUNK + tid;
    const int w = p & (W - 1);
    const int h = p / W;
    const uint32_t wm = wmask[b * W + w];
    const uint32_t fm = (fmcol[b * W + w] >> h) & 1u;

    float l[kC];
#pragma unroll
    for (int c = 0; c < kC; ++c) l[c] = L[c * CHUNK];
    float m = l[0];
#pragma unroll
    for (int c = 1; c < kC; ++c) m = fmaxf(m, l[c]);
    float s = 0.f;
#pragma unroll
    for (int c = 0; c < kC; ++c) s += __expf(l[c] - m);
    const float inv = __builtin_amdgcn_rcpf(s);
    const float lg = __logf(s);
    // argmax(mask): fm==0 -> ch0=1 -> 0; else first attention bit + 1; else 0
    const int target = (fm && wm) ? (1 + __builtin_ctz(wm)) : 0;
    cea += m + lg - l[target];  // -log p[target]
#pragma unroll
    for (int c = 0; c < kC; ++c) {
      const float pr = __expf(l[c] - m) * inv;
      s1a[c] += pr;
      const uint32_t mb = (c == 0) ? (fm ^ 1u) : (fm & ((wm >> (c - 1)) & 1u));
      ina[c] += mb ? pr : 0.f;
    }
    __syncthreads();  // all readers done before buffer reuse by TDM
  }

#pragma unroll
  for (int c = 0; c < kC; ++c) {
    atomicAdd(&red[c], s1a[c]);
    atomicAdd(&red[kC + c], ina[c]);
  }
  atomicAdd(&red[2 * kC], cea);
  __syncthreads();
  if (tid < kC) {
    s1_out[b * kC + tid] = red[tid];
    inter_out[b * kC + tid] = red[kC + tid];
  }
  if (tid == 0) ce_part[b] = red[2 * kC];
}

// ---------------------------------------------------------------------------
// Finalize: dice + CE means -> d_out[0]=loss_low, d_out[1]=loss_middle.
// ---------------------------------------------------------------------------
__global__ __launch_bounds__(512) void finalize_kernel(
    const float* __restrict__ s1_low, const float* __restrict__ s2_low,
    const float* __restrict__ in_low, const float* __restrict__ s1_mid,
    const float* __restrict__ s2_mid, const float* __restrict__ in_mid,
    const float* __restrict__ ce_low_part, const float* __restrict__ ce_mid_part,
    const int* __restrict__ length, float* __restrict__ out) {
  const int b = threadIdx.x;
  __shared__ float r0[kB], r1[kB], r2[kB], r3[kB];
  const float cel = ce_low_part[b];
  const float cem = ce_mid_part[b];
  const int len = length[b] - 1;
  float dl = 0.f, dm = 0.f;
  for (int c = 1; c < kC; ++c) {
    const float sl = (2.f * in_low[b * kC + c] + 1.f) /
                     (s1_low[b * kC + c] + s2_low[b * kC + c] + 1.f);
    const float sm = (2.f * in_mid[b * kC + c] + 1.f) /
                     (s1_mid[b * kC + c] + s2_mid[b * kC + c] + 1.f);
    const float v = (c <= len) ? 1.f : 0.f;
    dl += (1.f - sl) * v;
    dm += (1.f - sm) * v;
  }
  const float fl = (float)len;
  r0[b] = cel; r1[b] = cem; r2[b] = dl / fl; r3[b] = dm / fl;
  __syncthreads();
  for (int s = kB / 2; s > 0; s >>= 1) {
    if (b < s) {
      r0[b] += r0[b + s];
      r1[b] += r1[b + s];
      r2[b] += r2[b + s];
      r3[b] += r3[b + s];
    }
    __syncthreads();
  }
  if (b == 0) {
    out[0] = r0[0] / ((float)kB * (float)kHWLow) + r2[0] / (float)kB;  // loss_low
    out[1] = r1[0] / ((float)kB * (float)kHWMid) + r3[0] / (float)kB;  // loss_middle
  }
}

extern "C" void kernel_launch(void* const* d_in, const int* in_sizes, int n_in,
                              void* d_out, int out_size, void* d_ws, size_t ws_size,
                              hipStream_t stream) {
  (void)in_sizes; (void)n_in; (void)out_size; (void)ws_size;
  const float* cf_mid = (const float*)d_in[0];
  const float* cf_low = (const float*)d_in[1];
  const float* attn   = (const float*)d_in[2];
  const float* fore   = (const float*)d_in[3];
  const int*   length = (const int*)d_in[4];
  const float* alpha  = (const float*)d_in[5];
  float* out = (float*)d_out;

  char* ws = (char*)d_ws;
  uint32_t* wmask_low = (uint32_t*)ws; ws += (size_t)kB * kWLow * 4;
  uint32_t* wmask_mid = (uint32_t*)ws; ws += (size_t)kB * kWMid * 4;
  uint32_t* fmcol_low = (uint32_t*)ws; ws += (size_t)kB * kWLow * 4;
  uint32_t* fmcol_mid = (uint32_t*)ws; ws += (size_t)kB * kWMid * 4;
  float* s2_low    = (float*)ws; ws += (size_t)kB * kC * 4;
  float* s2_mid    = (float*)ws; ws += (size_t)kB * kC * 4;
  float* s1_low    = (float*)ws; ws += (size_t)kB * kC * 4;
  float* s1_mid    = (float*)ws; ws += (size_t)kB * kC * 4;
  float* inter_low = (float*)ws; ws += (size_t)kB * kC * 4;
  float* inter_mid = (float*)ws; ws += (size_t)kB * kC * 4;
  float* ce_low_part = (float*)ws; ws += (size_t)kB * 4;
  float* ce_mid_part = (float*)ws; ws += (size_t)kB * 4;

  {
    const int total = kB * (kWLow + kWMid);
    attn_mask_kernel<<<(total + 255) / 256, 256, 0, stream>>>(
        attn, length, alpha, wmask_low, wmask_mid);
  }
  fm_s2_kernel<<<kB, 128, 0, stream>>>(fore, wmask_low, wmask_mid, fmcol_low,
                                       fmcol_mid, s2_low, s2_mid);
  {
    const int total = kB * kC * kHLow * kWLow;
    mask_low_out_kernel<<<(total + 255) / 256, 256, 0, stream>>>(
        wmask_low, fmcol_low, out + 2);
  }
  char_loss_kernel<kHLow, kWLow, kHWLow / 256><<<kB, 256, 0, stream>>>(
      cf_low, wmask_low, fmcol_low, s1_low, inter_low, ce_low_part);
  char_loss_kernel<kHMid, kWMid, kHWMid / 256><<<kB, 256, 0, stream>>>(
      cf_mid, wmask_mid, fmcol_mid, s1_mid, inter_mid, ce_mid_part);
  finalize_kernel<<<1, kB, 0, stream>>>(s1_low, s2_low, inter_low, s1_mid,
                                        s2_mid, inter_mid, ce_low_part,
                                        ce_mid_part, length, out);
}